// NNarray_1271310320286
// MI455X (gfx1250) — compile-verified
//
#include <hip/hip_runtime.h>

typedef __attribute__((ext_vector_type(2))) float v2f;
typedef __attribute__((ext_vector_type(4))) float v4f;
typedef __attribute__((ext_vector_type(8))) float v8f;
typedef __attribute__((ext_vector_type(4))) unsigned int u32x4;
typedef __attribute__((ext_vector_type(8))) int i32x8;
typedef __attribute__((ext_vector_type(4))) int i32x4;

namespace {
constexpr float kDeltaV = 3.3f / 255.0f;
constexpr float kGmin   = 3.33e-07f;
constexpr float kDeltaG = 0.000333f - 3.33e-07f;
constexpr float kNoise  = 0.12f;

// LDS row strides in floats. Chosen so consecutive k-rows land 8 banks apart:
// the WMMA A/B fetch pattern has lanes 0-15 on row k and lanes 16-31 on row
// k+2 -> 16-lane groups are 16 banks apart -> conflict-free on 64 banks.
constexpr int GT_S = 72;   // GeffT: 64 rows (k) x 64 cols (m), padded 64->72
constexpr int XS_S = 136;  // Xs:    64 rows (k) x 128 cols (l), padded 128->136
}  // namespace

__global__ __launch_bounds__(256) void crxb_wmma_kernel(
    const float* __restrict__ x,    // [8,1,16,64,256]
    const float* __restrict__ w,    // [16,16,64,64]
    const float* __restrict__ nz,   // [2,16,16,64,64]
    float* __restrict__ out)        // [8,1024,16,256]
{
  __shared__ float GeffT[64 * GT_S];  // 18.0 KB
  __shared__ float Xs[64 * XS_S];     // 34.0 KB  (total 52 KB)

  const int col = blockIdx.x;      // 0..15
  const int row = blockIdx.y;      // 0..15
  const int b   = blockIdx.z >> 1; // 0..7
  const int h   = blockIdx.z & 1;  // L half: columns [h*128, h*128+128)

  const int t = threadIdx.x;       // 0..255 (8 wave32s)
  const int wave = t >> 5;

  // ---- Build effective conductance matrix, transposed, with DAC scale folded:
  //      GeffT[k][m] = dV * [ (Gmin + relu(W)dG)(1+.12 n0) - (Gmin + relu(-W)dG)(1+.12 n1) ]
  const int cr = col * 16 + row;
  const float* wp  = w  + (size_t)cr * 4096;
  const float* n0p = nz + (size_t)cr * 4096;
  const float* n1p = nz + (size_t)(256 + cr) * 4096;
#pragma unroll
  for (int i = 0; i < 16; ++i) {
    const int idx = i * 256 + t;         // coalesced 1KB bursts
    const int m = idx >> 6;              // output row of this crossbar
    const int k = idx & 63;              // contraction index
    const float wv = wp[idx];
    const float gp = fmaf(fmaxf(wv, 0.0f),  kDeltaG, kGmin) * fmaf(kNoise, n0p[idx], 1.0f);
    const float gn = fmaf(fmaxf(-wv, 0.0f), kDeltaG, kGmin) * fmaf(kNoise, n1p[idx], 1.0f);
    GeffT[k * GT_S + m] = (gp - gn) * kDeltaV;
  }

  // ---- Stage X chunk [64 rows x 128 cols] into LDS with +8-dword row pad.
  const float* xsrc = x + (((size_t)b * 16 + row) * 64) * 256 + (size_t)h * 128;

#if __has_builtin(__builtin_amdgcn_tensor_load_to_lds) && \
    __has_builtin(__builtin_amdgcn_s_wait_tensorcnt)
  // CDNA5 Tensor Data Mover: one wave issues a 2D tile DMA (128 x 64 elems,
  // 4B each, tensor row stride 256 elems). TDM's LDS padding (pad_interval=6
  // -> every 128 dwords, pad_amount=7 -> 8 dwords) reproduces XS_S = 136
  // floats/row, so the WMMA B-fetch stays bank-conflict free.
  if (wave == 0) {
    const unsigned long long ga = (unsigned long long)(uintptr_t)xsrc;
    const unsigned int lds_base = (unsigned int)(uintptr_t)(void*)Xs;  // addr[31:0] = LDS offset
    u32x4 g0 = {
        1u,                                                  // count=1 (valid user D#)
        lds_base,                                            // lds_addr
        (unsigned int)(ga & 0xFFFFFFFFu),                    // global_addr[31:0]
        (unsigned int)((ga >> 32) & 0x01FFFFFFu) | (2u << 30)  // addr[56:32] | type=2
    };
    i32x8 g1 = {
        (2 << 16) |            // data_size = 4 bytes
        (1 << 20) |            // pad_enable
        (6 << 22) |            // pad_interval: 128 dwords
        (7 << 25),             // pad_amount: 8 dwords
        (128 << 16),           // atomic_barrier_addr=0 | tensor_dim0(=128).lo16
        (64 << 16),            // tensor_dim0.hi16=0 | tensor_dim1(=64).lo16
        (128 << 16),           // tensor_dim1.hi16=0 | tile_dim0 = 128
        64,                    // tile_dim1 = 64 | tile_dim2 = 0 (2D)
        256,                   // tensor_dim0_stride.lo32 = 256 elems
        0,                     // tensor_dim0_stride.hi16 | tensor_dim1_stride.lo16
        0                      // tensor_dim1_stride.hi32
    };
    i32x4 gz4 = {0, 0, 0, 0};          // groups 2/3 unused for 2D tile
    i32x8 gz8 = {0, 0, 0, 0, 0, 0, 0, 0};
    __builtin_amdgcn_tensor_load_to_lds(g0, g1, gz4, gz4, gz8, 0);
    __builtin_amdgcn_s_wait_tensorcnt(0);  // TENSORcnt is per-wave: issuer waits
  }
#else
  // Fallback: cooperative float4 staging through VGPRs.
#pragma unroll
  for (int it = 0; it < 8; ++it) {
    const int p  = it * 256 + t;         // float4 index 0..2047
    const int k  = p >> 5;
    const int l4 = (p & 31) * 4;
    v4f v = *(const v4f*)(xsrc + (size_t)k * 256 + l4);
    *(v4f*)&Xs[k * XS_S + l4] = v;
  }
#endif
  __syncthreads();

  // ---- WMMA stage: wave -> one 16-row M tile x four 16-col N tiles.
  const int lane = t & 31;
  const int l15  = lane & 15;
  const int hi   = lane >> 4;            // half-wave select (k vs k+2 rows)
  const int mT   = (wave & 3) * 16;      // M tile base
  const int nB   = (wave >> 2) * 64;     // N group base (0 or 64 within chunk)

  // Preload A (16x64 tile of GeffT) into 32 VGPRs; reused for all 4 N tiles.
  // V_WMMA_F32_16X16X4_F32 A layout: lanes0-15 {K=k0,k0+1}, lanes16-31 {k0+2,k0+3}.
  v2f a[16];
#pragma unroll
  for (int kk = 0; kk < 16; ++kk) {
    const float* gp = &GeffT[(kk * 4 + hi * 2) * GT_S + mT + l15];
    a[kk].x = gp[0];
    a[kk].y = gp[GT_S];
  }

  // Four independent accumulator chains to pipeline the matrix unit.
  v8f acc0 = {}, acc1 = {}, acc2 = {}, acc3 = {};
#pragma unroll
  for (int kk = 0; kk < 16; ++kk) {
    // B layout (4x16): VGPR0 = {K=k0 | k0+2}, VGPR1 = {k0+1 | k0+3}, N = lane&15.
    const float* xp = &Xs[(kk * 4 + hi * 2) * XS_S + nB + l15];
    v2f b0, b1, b2, b3;
    b0.x = xp[0];   b0.y = xp[XS_S];
    b1.x = xp[16];  b1.y = xp[XS_S + 16];
    b2.x = xp[32];  b2.y = xp[XS_S + 32];
    b3.x = xp[48];  b3.y = xp[XS_S + 48];
    acc0 = __builtin_amdgcn_wmma_f32_16x16x4_f32(false, a[kk], false, b0, (short)0, acc0, false, false);
    acc1 = __builtin_amdgcn_wmma_f32_16x16x4_f32(false, a[kk], false, b1, (short)0, acc1, false, false);
    acc2 = __builtin_amdgcn_wmma_f32_16x16x4_f32(false, a[kk], false, b2, (short)0, acc2, false, false);
    acc3 = __builtin_amdgcn_wmma_f32_16x16x4_f32(false, a[kk], false, b3, (short)0, acc3, false, false);
  }

  // ---- Store. out flat index: ((b*1024 + col*64 + m)*16 + row)*256 + l.
  // C/D layout: VGPR v holds rows m=v (lanes 0-15) and m=v+8 (lanes 16-31).
  float* ob = out + (size_t)b * 4194304 + (size_t)col * 262144 +
              (size_t)row * 256 + (size_t)h * 128 + nB + l15;
#pragma unroll
  for (int v = 0; v < 8; ++v) {
    const int m = mT + v + hi * 8;
    float* o = ob + (size_t)m * 4096;
    o[0]  = acc0[v];
    o[16] = acc1[v];
    o[32] = acc2[v];
    o[48] = acc3[v];
  }
}

extern "C" void kernel_launch(void* const* d_in, const int* in_sizes, int n_in,
                              void* d_out, int out_size, void* d_ws, size_t ws_size,
                              hipStream_t stream) {
  const float* x  = (const float*)d_in[0];   // inputs [8,1,16,64,256]
  const float* w  = (const float*)d_in[1];   // weight [16,16,64,64]
  const float* nz = (const float*)d_in[2];   // noise  [2,16,16,64,64]
  float* out = (float*)d_out;                // [8,1024,16,256]

  dim3 grid(16, 16, 16);  // (col, row, b*2 + L-half)
  crxb_wmma_kernel<<<grid, dim3(256), 0, stream>>>(x, w, nz, out);
}